// DecModel_16209206575260
// MI455X (gfx1250) — compile-verified
//
#include <hip/hip_runtime.h>
#include <math.h>

// ---------------------------------------------------------------------------
// DecModel fused kernel for MI455X (gfx1250, wave32, WMMA f32 16x16x4)
// B=1024, NV=16, NS=32, P=8, BASE_DIM=32, OUT_DIM=13
// One block = 128 threads (4 waves) = 16 batch elements. Grid = B/16 = 64.
// ---------------------------------------------------------------------------

typedef float v2f __attribute__((ext_vector_type(2)));
typedef float v8f __attribute__((ext_vector_type(8)));

#define LN_EPS_F 1e-6f

__device__ __forceinline__ float grp16_sum(float x) {
  // sum across the 16-lane half-group containing this lane (wave32)
  x += __shfl_xor(x, 1, 32);
  x += __shfl_xor(x, 2, 32);
  x += __shfl_xor(x, 4, 32);
  x += __shfl_xor(x, 8, 32);
  return x;
}

__device__ __forceinline__ void qrotf(float qx, float qy, float qz, float qw,
                                      float vx, float vy, float vz,
                                      float& rx, float& ry, float& rz) {
  // t = 2*cross(qv, v); r = v + qw*t + cross(qv, t)
  float tx = 2.f * (qy * vz - qz * vy);
  float ty = 2.f * (qz * vx - qx * vz);
  float tz = 2.f * (qx * vy - qy * vx);
  rx = vx + qw * tx + (qy * tz - qz * ty);
  ry = vy + qw * ty + (qz * tx - qx * tz);
  rz = vz + qw * tz + (qx * ty - qy * tx);
}

__global__ __launch_bounds__(128) void decmodel_kernel(
    const float* __restrict__ pos, const float* __restrict__ quat,
    const float* __restrict__ aabb, const float* __restrict__ xz,
    const int* __restrict__ widx, const float* __restrict__ plsp,
    const float* __restrict__ W1, const float* __restrict__ b1,
    const float* __restrict__ g1, const float* __restrict__ be1,
    const float* __restrict__ W2, const float* __restrict__ b2,
    const float* __restrict__ g2, const float* __restrict__ be2,
    const float* __restrict__ W3, const float* __restrict__ b3,
    const float* __restrict__ g3, const float* __restrict__ be3,
    const float* __restrict__ W4, const float* __restrict__ b4,
    float* __restrict__ out) {
  // LDS
  __shared__ float featW[4][32 * 26];  // per-wave 32 rows x 24 feats (stride 26)
  __shared__ float zbuf[16 * 32];      // z_pair for the block's 16 batch elems
  __shared__ float z2w[4][16 * 32];    // per-wave staging for layers 3/4 A-mats

  const int tid = threadIdx.x;
  const int wave = tid >> 5;
  const int lane = tid & 31;
  const int half = lane >> 4;    // 0: lanes 0-15, 1: lanes 16-31
  const int n = lane & 15;       // column within a 16-wide tile
  const int koff = half * 2;     // K offset of this lane's A/B fragment
  const int b_base = blockIdx.x * 16;

  // -------------------- preload W1 B-fragments (23x16, K padded to 24) -----
  v2f bf1[6];
#pragma unroll
  for (int j = 0; j < 6; ++j) {
    int kk = j * 4 + koff;
    bf1[j].x = (kk < 23) ? W1[kk * 16 + n] : 0.f;
    bf1[j].y = (kk + 1 < 23) ? W1[(kk + 1) * 16 + n] : 0.f;
  }
  const float b1n = b1[n], g1n = g1[n], be1n = be1[n];
  const float escale = expf(plsp[0]);

  // ========================== Stage A: per (b, d) ==========================
  float* fw = featW[wave];
  for (int bl = 0; bl < 4; ++bl) {
    const int b = b_base + wave * 4 + bl;

    // ---- relative pose (redundant per-lane; ~100 flops) ----
    const float p1x = pos[b * 6 + 0], p1y = pos[b * 6 + 1], p1z = pos[b * 6 + 2];
    const float p2x = pos[b * 6 + 3], p2y = pos[b * 6 + 4], p2z = pos[b * 6 + 5];
    const float ax = -quat[b * 8 + 0], ay = -quat[b * 8 + 1],
                az = -quat[b * 8 + 2], aw = quat[b * 8 + 3];  // q1 conj
    const float cx = quat[b * 8 + 4], cy = quat[b * 8 + 5],
                cz = quat[b * 8 + 6], cw = quat[b * 8 + 7];   // q2

    float r1x, r1y, r1z, r2x, r2y, r2z;
    qrotf(ax, ay, az, aw, p1x, p1y, p1z, r1x, r1y, r1z);
    qrotf(ax, ay, az, aw, p2x, p2y, p2z, r2x, r2y, r2z);
    const float rpx = r2x - r1x, rpy = r2y - r1y, rpz = r2z - r1z;

    // relquat = qmul(q1i, q2)
    const float rqw = aw * cw - (ax * cx + ay * cy + az * cz);
    const float rqx = aw * cx + cw * ax + (ay * cz - az * cy);
    const float rqy = aw * cy + cw * ay + (az * cx - ax * cz);
    const float rqz = aw * cz + cw * az + (ax * cy - ay * cx);

    float pf[12];
    pf[0] = rpx * escale; pf[1] = rpy * escale; pf[2] = rpz * escale;
    pf[3] = 1.f - 2.f * (rqy * rqy + rqz * rqz);
    pf[4] = 2.f * (rqx * rqy - rqz * rqw);
    pf[5] = 2.f * (rqx * rqz + rqy * rqw);
    pf[6] = 2.f * (rqx * rqy + rqz * rqw);
    pf[7] = 1.f - 2.f * (rqx * rqx + rqz * rqz);
    pf[8] = 2.f * (rqy * rqz - rqx * rqw);
    pf[9] = 2.f * (rqx * rqz - rqy * rqw);
    pf[10] = 2.f * (rqy * rqz + rqx * rqw);
    pf[11] = 1.f - 2.f * (rqx * rqx + rqy * rqy);

    for (int d = 0; d < 2; ++d) {
      const float lox = aabb[b * 12 + d * 6 + 0];
      const float loy = aabb[b * 12 + d * 6 + 1];
      const float loz = aabb[b * 12 + d * 6 + 2];
      const float hix = aabb[b * 12 + d * 6 + 3];
      const float hiy = aabb[b * 12 + d * 6 + 4];
      const float hiz = aabb[b * 12 + d * 6 + 5];

      // lane == sample index s (NS == 32 == wave size)
      const int w = widx[(b * 2 + d) * 32 + lane];
      const float gi = (float)(w >> 8);
      const float gj = (float)((w >> 4) & 15);
      const float gk = (float)(w & 15);
      float px = (gi + 0.5f) * (1.f / 16.f) * (hix - lox) + lox;
      float py = (gj + 0.5f) * (1.f / 16.f) * (hiy - loy) + loy;
      float pz = (gk + 0.5f) * (1.f / 16.f) * (hiz - loz) + loz;
      if (d == 1) {
        float tx, ty, tz;
        qrotf(rqx, rqy, rqz, rqw, px, py, pz, tx, ty, tz);
        px = tx + rpx; py = ty + rpy; pz = tz + rpz;
      }

      const float* zp = xz + ((size_t)(b * 2 + d) * 4096 + (size_t)w) * 8;
      float* frow = fw + lane * 26;
      frow[0] = px; frow[1] = py; frow[2] = pz;
#pragma unroll
      for (int p = 0; p < 8; ++p) frow[3 + p] = zp[p];
      if (d) {
#pragma unroll
        for (int p = 0; p < 12; ++p) frow[11 + p] = pf[p];
      } else {
#pragma unroll
        for (int p = 0; p < 12; ++p) frow[11 + p] = 0.f;
      }
      frow[23] = 0.f;  // K padding
      __syncthreads();

      // ---- layer 1: (32x24) @ (24x16) via 2 M-tiles x 6 K-steps of WMMA ---
      float acc = 0.f;  // per-lane column sum of 2*LN(relu(.)) over my rows
#pragma unroll
      for (int t = 0; t < 2; ++t) {
        v8f c = {b1n, b1n, b1n, b1n, b1n, b1n, b1n, b1n};
        const float* row = fw + (t * 16 + n) * 26;
#pragma unroll
        for (int j = 0; j < 6; ++j) {
          const int kk = j * 4 + koff;
          v2f a;
          a.x = row[kk];
          a.y = row[kk + 1];
          c = __builtin_amdgcn_wmma_f32_16x16x4_f32(
              false, a, false, bf1[j], (short)0, c, false, false);
        }
        // relu + layernorm over the 16 outputs of each row, then *2, row-sum
#pragma unroll
        for (int v = 0; v < 8; ++v) {
          float x = fmaxf(c[v], 0.f);
          float s1 = grp16_sum(x);
          float s2 = grp16_sum(x * x);
          float m = s1 * (1.f / 16.f);
          float var = s2 * (1.f / 16.f) - m * m;
          float ln = (x - m) * rsqrtf(var + LN_EPS_F) * g1n + be1n;
          acc += 2.f * ln;  // z = z + z
        }
      }
      acc += __shfl_xor(acc, 16, 32);  // combine the two half-groups' rows
      if (lane < 16) zbuf[(wave * 4 + bl) * 32 + d * 16 + lane] = acc;
      __syncthreads();
    }
  }

  // ================= Stage B: 16x32 MLP tail (all waves, redundant) ========
  float* zw = z2w[wave];
  const float g2a = g2[n], g2b = g2[16 + n], be2a = be2[n], be2b = be2[16 + n];
  const float g3a = g3[n], g3b = g3[16 + n], be3a = be3[n], be3b = be3[16 + n];

  // ---- layer 2: X(16x32) @ W2(32x32) ----
  const float bb20 = b2[n], bb21 = b2[16 + n];
  v8f c0 = {bb20, bb20, bb20, bb20, bb20, bb20, bb20, bb20};
  v8f c1 = {bb21, bb21, bb21, bb21, bb21, bb21, bb21, bb21};
#pragma unroll
  for (int j = 0; j < 8; ++j) {
    const int kk = j * 4 + koff;
    v2f a;
    a.x = zbuf[n * 32 + kk];
    a.y = zbuf[n * 32 + kk + 1];
    v2f bA, bB;
    bA.x = W2[kk * 32 + n];        bA.y = W2[(kk + 1) * 32 + n];
    bB.x = W2[kk * 32 + 16 + n];   bB.y = W2[(kk + 1) * 32 + 16 + n];
    c0 = __builtin_amdgcn_wmma_f32_16x16x4_f32(false, a, false, bA, (short)0, c0, false, false);
    c1 = __builtin_amdgcn_wmma_f32_16x16x4_f32(false, a, false, bB, (short)0, c1, false, false);
  }
  v8f z0, z1, skip0, skip1;
#pragma unroll
  for (int v = 0; v < 8; ++v) {
    float x0 = fmaxf(c0[v], 0.f), x1 = fmaxf(c1[v], 0.f);
    float s1 = grp16_sum(x0 + x1);
    float s2 = grp16_sum(x0 * x0 + x1 * x1);
    float m = s1 * (1.f / 32.f);
    float var = s2 * (1.f / 32.f) - m * m;
    float rs = rsqrtf(var + LN_EPS_F);
    z0[v] = (x0 - m) * rs * g2a + be2a;
    z1[v] = (x1 - m) * rs * g2b + be2b;
  }
  skip0 = z0; skip1 = z1;
#pragma unroll
  for (int v = 0; v < 8; ++v) {
    zw[(v + half * 8) * 32 + n] = z0[v];
    zw[(v + half * 8) * 32 + 16 + n] = z1[v];
  }
  __syncthreads();

  // ---- layer 3: Z2(16x32) @ W3(32x32), then + skip ----
  const float bb30 = b3[n], bb31 = b3[16 + n];
  v8f d0 = {bb30, bb30, bb30, bb30, bb30, bb30, bb30, bb30};
  v8f d1 = {bb31, bb31, bb31, bb31, bb31, bb31, bb31, bb31};
#pragma unroll
  for (int j = 0; j < 8; ++j) {
    const int kk = j * 4 + koff;
    v2f a;
    a.x = zw[n * 32 + kk];
    a.y = zw[n * 32 + kk + 1];
    v2f bA, bB;
    bA.x = W3[kk * 32 + n];        bA.y = W3[(kk + 1) * 32 + n];
    bB.x = W3[kk * 32 + 16 + n];   bB.y = W3[(kk + 1) * 32 + 16 + n];
    d0 = __builtin_amdgcn_wmma_f32_16x16x4_f32(false, a, false, bA, (short)0, d0, false, false);
    d1 = __builtin_amdgcn_wmma_f32_16x16x4_f32(false, a, false, bB, (short)0, d1, false, false);
  }
#pragma unroll
  for (int v = 0; v < 8; ++v) {
    float x0 = fmaxf(d0[v], 0.f), x1 = fmaxf(d1[v], 0.f);
    float s1 = grp16_sum(x0 + x1);
    float s2 = grp16_sum(x0 * x0 + x1 * x1);
    float m = s1 * (1.f / 32.f);
    float var = s2 * (1.f / 32.f) - m * m;
    float rs = rsqrtf(var + LN_EPS_F);
    z0[v] = (x0 - m) * rs * g3a + be3a + skip0[v];
    z1[v] = (x1 - m) * rs * g3b + be3b + skip1[v];
  }
  __syncthreads();  // all lanes done reading zw before overwrite
#pragma unroll
  for (int v = 0; v < 8; ++v) {
    zw[(v + half * 8) * 32 + n] = z0[v];
    zw[(v + half * 8) * 32 + 16 + n] = z1[v];
  }
  __syncthreads();

  // ---- layer 4: Z(16x32) @ W4(32x13 padded to 16), tanh ----
  const float bb4 = (n < 13) ? b4[n] : 0.f;
  v8f c4 = {bb4, bb4, bb4, bb4, bb4, bb4, bb4, bb4};
#pragma unroll
  for (int j = 0; j < 8; ++j) {
    const int kk = j * 4 + koff;
    v2f a;
    a.x = zw[n * 32 + kk];
    a.y = zw[n * 32 + kk + 1];
    v2f bw;
    bw.x = (n < 13) ? W4[kk * 13 + n] : 0.f;
    bw.y = (n < 13) ? W4[(kk + 1) * 13 + n] : 0.f;
    c4 = __builtin_amdgcn_wmma_f32_16x16x4_f32(false, a, false, bw, (short)0, c4, false, false);
  }
  if (wave == 0 && n < 13) {
#pragma unroll
    for (int v = 0; v < 8; ++v) {
      out[(size_t)(b_base + v + half * 8) * 13 + n] = tanhf(c4[v]);
    }
  }
}

extern "C" void kernel_launch(void* const* d_in, const int* in_sizes, int n_in,
                              void* d_out, int out_size, void* d_ws, size_t ws_size,
                              hipStream_t stream) {
  const float* pos  = (const float*)d_in[0];
  const float* quat = (const float*)d_in[1];
  const float* aabb = (const float*)d_in[2];
  const float* xz   = (const float*)d_in[3];
  const int*   widx = (const int*)d_in[4];
  const float* pls  = (const float*)d_in[5];
  const float* W1 = (const float*)d_in[6];
  const float* b1 = (const float*)d_in[7];
  const float* g1 = (const float*)d_in[8];
  const float* be1 = (const float*)d_in[9];
  const float* W2 = (const float*)d_in[10];
  const float* b2 = (const float*)d_in[11];
  const float* g2 = (const float*)d_in[12];
  const float* be2 = (const float*)d_in[13];
  const float* W3 = (const float*)d_in[14];
  const float* b3 = (const float*)d_in[15];
  const float* g3 = (const float*)d_in[16];
  const float* be3 = (const float*)d_in[17];
  const float* W4 = (const float*)d_in[18];
  const float* b4 = (const float*)d_in[19];
  float* out = (float*)d_out;

  const int Bn = in_sizes[0] / 6;   // pos is (B,2,3)
  const int grid = Bn / 16;         // 16 batch elements per block
  hipLaunchKernelGGL(decmodel_kernel, dim3(grid), dim3(128), 0, stream,
                     pos, quat, aabb, xz, widx, pls,
                     W1, b1, g1, be1, W2, b2, g2, be2,
                     W3, b3, g3, be3, W4, b4, out);
}